// Olmo3MoeRouter_23141283791733
// MI455X (gfx1250) — compile-verified
//
#include <hip/hip_runtime.h>
#include <math.h>

typedef float v2f __attribute__((ext_vector_type(2)));
typedef float v8f __attribute__((ext_vector_type(8)));

#define HIDDEN   2048
#define N_EXPERTS 128
#define TOPK     8
#define TOK_TILE 16
#define LDS_STRIDE 132   // 128 + 4 pad: splits bank sets between lane halves

__global__ __launch_bounds__(256) void olmo3_router_kernel(
    const float* __restrict__ x,      // [32768, 2048]
    const float* __restrict__ gw,     // [128, 2048]
    float* __restrict__ w_out,        // [32768, 8]
    int*   __restrict__ i_out)        // [32768, 8]
{
    __shared__ float s_logits[TOK_TILE * LDS_STRIDE];

    const int wave    = threadIdx.x >> 5;   // 0..7  -> expert tile
    const int lane    = threadIdx.x & 31;
    const int lane_lo = lane & 15;
    const int half    = lane >> 4;          // 0: K pair (k,k+1), 1: (k+2,k+3)
    const int t0      = blockIdx.x * TOK_TILE;

    // ---- GEMM: 16 tokens x 16 experts per wave, K = 2048 via wmma f32 16x16x4
    const float* xrow = x  + (size_t)(t0 + lane_lo) * HIDDEN + 2 * half;
    const float* grow = gw + (size_t)(wave * 16 + lane_lo) * HIDDEN + 2 * half;

    v8f c0 = {};   // even k-steps
    v8f c1 = {};   // odd  k-steps (breaks accumulator dependency chain)

    #pragma unroll 4
    for (int k = 0; k < HIDDEN; k += 8) {
        v2f a0 = *(const v2f*)(xrow + k);
        v2f b0 = *(const v2f*)(grow + k);
        v2f a1 = *(const v2f*)(xrow + k + 4);
        v2f b1 = *(const v2f*)(grow + k + 4);
        c0 = __builtin_amdgcn_wmma_f32_16x16x4_f32(
                false, a0, false, b0, (short)0, c0, false, false);
        c1 = __builtin_amdgcn_wmma_f32_16x16x4_f32(
                false, a1, false, b1, (short)0, c1, false, false);
    }

    // C layout: VGPR r -> row (r + 8*half), col lane_lo  (ISA 7.12.2)
    #pragma unroll
    for (int r = 0; r < 8; ++r) {
        int row = r + 8 * half;
        s_logits[row * LDS_STRIDE + wave * 16 + lane_lo] = c0[r] + c1[r];
    }

    __syncthreads();

    // ---- Top-8 + renormalized softmax weights.
    // softmax is monotonic -> top-k on logits; denominators cancel:
    //   w_i = exp(l_i - l_max) / sum_top8 exp(l_j - l_max)
    // wave w handles tokens 2w and 2w+1; lane holds 4 of 128 logits.
    #pragma unroll
    for (int tt = 0; tt < 2; ++tt) {
        const int tok = wave * 2 + tt;
        float v0 = s_logits[tok * LDS_STRIDE + lane];
        float v1 = s_logits[tok * LDS_STRIDE + lane + 32];
        float v2 = s_logits[tok * LDS_STRIDE + lane + 64];
        float v3 = s_logits[tok * LDS_STRIDE + lane + 96];

        float topv[TOPK];
        int   topi[TOPK];

        #pragma unroll
        for (int r = 0; r < TOPK; ++r) {
            // local argmax over the 4 per-lane candidates (lower index wins ties)
            float bv = v0; int bi = lane;
            if (v1 > bv) { bv = v1; bi = lane + 32; }
            if (v2 > bv) { bv = v2; bi = lane + 64; }
            if (v3 > bv) { bv = v3; bi = lane + 96; }
            // wave32 butterfly argmax; tie -> lower expert index (matches lax.top_k)
            #pragma unroll
            for (int off = 16; off >= 1; off >>= 1) {
                float ov = __shfl_xor(bv, off, 32);
                int   oi = __shfl_xor(bi, off, 32);
                if (ov > bv || (ov == bv && oi < bi)) { bv = ov; bi = oi; }
            }
            topv[r] = bv;
            topi[r] = bi;
            // winning lane retires its element
            if (bi == lane)       v0 = -INFINITY;
            if (bi == lane + 32)  v1 = -INFINITY;
            if (bi == lane + 64)  v2 = -INFINITY;
            if (bi == lane + 96)  v3 = -INFINITY;
        }

        if (lane == 0) {
            const float m = topv[0];
            float e[TOPK];
            float sum = 0.f;
            #pragma unroll
            for (int r = 0; r < TOPK; ++r) { e[r] = expf(topv[r] - m); sum += e[r]; }
            const float inv = 1.f / sum;
            const size_t base = (size_t)(t0 + tok) * TOPK;
            #pragma unroll
            for (int r = 0; r < TOPK; ++r) {
                w_out[base + r] = e[r] * inv;
                i_out[base + r] = topi[r];
            }
        }
    }
}

extern "C" void kernel_launch(void* const* d_in, const int* in_sizes, int n_in,
                              void* d_out, int out_size, void* d_ws, size_t ws_size,
                              hipStream_t stream) {
    const float* x  = (const float*)d_in[0];   // [32768, 2048] fp32
    const float* gw = (const float*)d_in[1];   // [128, 2048] fp32

    const int tokens = in_sizes[0] / HIDDEN;   // 32768

    // d_out: expert_weights [tokens,8] fp32, then expert_indices [tokens,8] int32
    float* w_out = (float*)d_out;
    int*   i_out = (int*)((float*)d_out + (size_t)tokens * TOPK);

    dim3 grid(tokens / TOK_TILE);   // 2048 workgroups, one 16-token tile each
    dim3 block(256);                // 8 waves: one expert tile of 16 per wave
    olmo3_router_kernel<<<grid, block, 0, stream>>>(x, gw, w_out, i_out);
}